// KoolModel_77223511982152
// MI455X (gfx1250) — compile-verified
//
#include <hip/hip_runtime.h>

// Problem sizes (compile-time constants from the reference).
#define L_  1024
#define B_  256
#define D_  256
#define T_  50
#define H_  8
#define DH_ 32

// CDNA5 async global->LDS copy path (ASYNCcnt, CDNA5 ISA).
// If the assembler rejects the mnemonic, flip to 0 (plain load + ds_store).
#define USE_ASYNC_LDS 1

typedef __attribute__((ext_vector_type(2))) float v2f;
typedef __attribute__((ext_vector_type(8))) float v8f;

// ---------------------------------------------------------------------------
// Generic single-wave 16x16 f32 WMMA tile GEMM:  C[16x16] = scale * A(16xK)·B(Kx16)
// A layout (ISA 7.12.2): lane m=(lane&15); lanes>=16 hold K+2.
// B/C: row striped across lanes. TA/TB transpose flags; mValid/nValid pad H=8.
// ---------------------------------------------------------------------------
template <bool TA, bool TB>
__device__ __forceinline__ void wmma_tile(const float* __restrict__ A, int lda,
                                          const float* __restrict__ B, int ldb,
                                          float* __restrict__ C, int ldc,
                                          int K, float scale, int mValid, int nValid) {
  const int lane = threadIdx.x & 31;
  const int half = lane >> 4;
  const int mn   = lane & 15;
  v8f acc = {0.f, 0.f, 0.f, 0.f, 0.f, 0.f, 0.f, 0.f};
  for (int k0 = 0; k0 < K; k0 += 4) {
    const int ka = k0 + half * 2;
    v2f a, b;
    if (mn < mValid) {
      if (TA) { a.x = A[(size_t)ka * lda + mn];   a.y = A[(size_t)(ka + 1) * lda + mn]; }
      else    { a.x = A[(size_t)mn * lda + ka];   a.y = A[(size_t)mn * lda + ka + 1];   }
    } else { a.x = 0.f; a.y = 0.f; }
    if (mn < nValid) {
      if (TB) { b.x = B[(size_t)mn * ldb + ka];   b.y = B[(size_t)mn * ldb + ka + 1];   }
      else    { b.x = B[(size_t)ka * ldb + mn];   b.y = B[(size_t)(ka + 1) * ldb + mn]; }
    } else { b.x = 0.f; b.y = 0.f; }
    acc = __builtin_amdgcn_wmma_f32_16x16x4_f32(false, a, false, b, (short)0, acc,
                                                false, false);
  }
  if (mn < nValid) {
#pragma unroll
    for (int r = 0; r < 8; ++r) {
      const int m = half * 8 + r;
      if (m < mValid) C[(size_t)m * ldc + mn] = acc[r] * scale;
    }
  }
}

// --------------------------- reductions ------------------------------------
__device__ __forceinline__ float wave_sum(float v) {
  for (int o = 16; o; o >>= 1) v += __shfl_xor(v, o, 32);
  return v;
}
__device__ __forceinline__ float wave_max(float v) {
  for (int o = 16; o; o >>= 1) v = fmaxf(v, __shfl_xor(v, o, 32));
  return v;
}

// ---------------------------------------------------------------------------
// K1a: 8-way split partial mean over L (2048 blocks, 4 independent f32 chains)
// ---------------------------------------------------------------------------
__global__ void k_mean_partial(const float* __restrict__ ebp, float* __restrict__ psum) {
  const int c = blockIdx.x;           // L-chunk 0..7 (128 rows each)
  const int b = blockIdx.y;
  const int d = threadIdx.x;
  const float* p = ebp + (size_t)(c * 128) * (B_ * D_) + (size_t)b * D_ + d;
  float s0 = 0.f, s1 = 0.f, s2 = 0.f, s3 = 0.f;
  for (int l = 0; l < 128; l += 4) {
    s0 += p[(size_t)(l + 0) * (B_ * D_)];
    s1 += p[(size_t)(l + 1) * (B_ * D_)];
    s2 += p[(size_t)(l + 2) * (B_ * D_)];
    s3 += p[(size_t)(l + 3) * (B_ * D_)];
  }
  psum[((size_t)c * B_ + b) * D_ + d] = (s0 + s1) + (s2 + s3);
}

// K1b: finalize context = [hbar, ebp[pa[:,-1]], ebp[pa[:,0]]]  (B x 768)
__global__ void k_context_fin(const float* __restrict__ ebp, const float* __restrict__ psum,
                              const int* __restrict__ pa, float* __restrict__ context) {
  const int b = blockIdx.x, d = threadIdx.x;
  float s = 0.f;
  for (int c = 0; c < 8; ++c) s += psum[((size_t)c * B_ + b) * D_ + d];
  const int last = pa[b * T_ + (T_ - 1)];
  const int first = pa[b * T_];
  context[(size_t)b * (3 * D_) + d]          = s * (1.0f / L_);
  context[(size_t)b * (3 * D_) + D_ + d]     = ebp[(size_t)last  * (B_ * D_) + (size_t)b * D_ + d];
  context[(size_t)b * (3 * D_) + 2 * D_ + d] = ebp[(size_t)first * (B_ * D_) + (size_t)b * D_ + d];
}

// ----------------- generic 1-wave-per-tile GEMM wrappers -------------------
__global__ void k_gemm_nn(const float* __restrict__ A, int lda, const float* __restrict__ Bm,
                          int ldb, float* __restrict__ C, int ldc, int K, float scale) {
  const int m0 = blockIdx.x * 16, n0 = blockIdx.y * 16;
  wmma_tile<false, false>(A + (size_t)m0 * lda, lda, Bm + n0, ldb,
                          C + (size_t)m0 * ldc + n0, ldc, K, scale, 16, 16);
}
__global__ void k_gemm_nt(const float* __restrict__ A, int lda, const float* __restrict__ Bt,
                          int ldb, float* __restrict__ C, int ldc, int K, float scale) {
  const int m0 = blockIdx.x * 16, n0 = blockIdx.y * 16;
  wmma_tile<false, true>(A + (size_t)m0 * lda, lda, Bt + (size_t)n0 * ldb, ldb,
                         C + (size_t)m0 * ldc + n0, ldc, K, scale, 16, 16);
}

// K3: P1[b,h,d] = (1/sqrt(32)) * sum_j q1[b,h*32+j] * Wk1[d,h*32+j]
__global__ void k_p1(const float* __restrict__ q1, const float* __restrict__ Wk1,
                     float* __restrict__ P1) {
  const int b0 = blockIdx.x * 16, d0 = blockIdx.y * 16, h = blockIdx.z;
  wmma_tile<false, true>(q1 + (size_t)b0 * D_ + h * DH_, D_,
                         Wk1 + (size_t)d0 * D_ + h * DH_, D_,
                         P1 + (size_t)b0 * (H_ * D_) + h * D_ + d0, H_ * D_,
                         DH_, 0.1767766952966369f, 16, 16);
}

// ---------------------------------------------------------------------------
// K4 (streams ebp at HBM rate): sc[l,b,h] = sum_d ebp[l,b,d] * P1[b,h,d]
// LDS-staged: coalesced (async) global->LDS row loads, WMMA A-fragments from
// LDS (ds_load_b64, conflict-free 66-float pitch). 256 thr = 8 waves, each
// wave owns one 16-row l-tile; K processed in 64-float chunks.
// ---------------------------------------------------------------------------
#define SC_LROWS 128
#define SC_KC    64
#define SC_PITA  66
#define SC_PITB  258

__global__ void __launch_bounds__(256) k_scores(const float* __restrict__ ebp,
                                                const float* __restrict__ P1,
                                                float* __restrict__ sc) {
  const int l0 = blockIdx.x * SC_LROWS;
  const int b  = blockIdx.y;
  __shared__ float tile[SC_LROWS][SC_PITA];   // 33792 B
  __shared__ float p1s[H_][SC_PITB];          //  8256 B

  const int tid = threadIdx.x;
  for (int i = tid; i < H_ * D_; i += 256) {  // stage P1 (8 KB) once, coalesced
    p1s[i >> 8][i & 255] = P1[(size_t)b * (H_ * D_) + i];
  }

  const int lane = tid & 31;
  const int wv   = tid >> 5;
  const int half = lane >> 4;
  const int mn   = lane & 15;
  v8f acc = {0.f, 0.f, 0.f, 0.f, 0.f, 0.f, 0.f, 0.f};

  for (int c = 0; c < D_ / SC_KC; ++c) {
    __syncthreads();                          // previous chunk fully consumed
#pragma unroll
    for (int i = 0; i < 8; ++i) {             // 2048 float4 = 128 rows x 256 B
      const int idx = tid + i * 256;
      const int r   = idx >> 4;               // row 0..127
      const int c4  = idx & 15;               // float4 within row-chunk
      const float* gsrc = ebp + (size_t)(l0 + r) * (B_ * D_) + (size_t)b * D_
                              + c * SC_KC + c4 * 4;
#if USE_ASYNC_LDS
      const unsigned ldsAddr = (unsigned)(uintptr_t)&tile[r][c4 * 4];
      asm volatile("global_load_async_to_lds_b128 %0, %1, off"
                   :: "v"(ldsAddr), "v"(gsrc) : "memory");
#else
      *(float4*)&tile[r][c4 * 4] = *(const float4*)gsrc;
#endif
    }
#if USE_ASYNC_LDS
    asm volatile("s_wait_asynccnt 0x0" ::: "memory");
#endif
    __syncthreads();
#pragma unroll
    for (int k0 = 0; k0 < SC_KC; k0 += 4) {
      const int ka = k0 + half * 2;
      v2f a, bf;
      a.x = tile[wv * 16 + mn][ka];
      a.y = tile[wv * 16 + mn][ka + 1];
      if (mn < H_) {
        bf.x = p1s[mn][c * SC_KC + ka];
        bf.y = p1s[mn][c * SC_KC + ka + 1];
      } else { bf.x = 0.f; bf.y = 0.f; }
      acc = __builtin_amdgcn_wmma_f32_16x16x4_f32(false, a, false, bf, (short)0, acc,
                                                  false, false);
    }
  }
  if (mn < H_) {
#pragma unroll
    for (int r = 0; r < 8; ++r) {
      const int m = half * 8 + r;
      sc[(size_t)(l0 + wv * 16 + m) * (B_ * H_) + b * H_ + mn] = acc[r];
    }
  }
}

// K5: masked softmax over L for each (b,h), in place on sc.
__global__ void k_softmax_sc(float* __restrict__ sc, const int* __restrict__ pa) {
  const int b = blockIdx.x, h = blockIdx.y;
  __shared__ float v[L_];
  __shared__ float red[8];
  float* col = sc + (size_t)b * H_ + h;
  for (int i = threadIdx.x; i < L_; i += blockDim.x) v[i] = col[(size_t)i * (B_ * H_)];
  __syncthreads();
  for (int t = threadIdx.x; t < T_; t += blockDim.x) v[pa[b * T_ + t]] = -__builtin_inff();
  __syncthreads();
  const int wid = threadIdx.x >> 5;
  float m = -__builtin_inff();
  for (int i = threadIdx.x; i < L_; i += blockDim.x) m = fmaxf(m, v[i]);
  m = wave_max(m);
  if ((threadIdx.x & 31) == 0) red[wid] = m;
  __syncthreads();
  m = red[0];
  for (int w = 1; w < 8; ++w) m = fmaxf(m, red[w]);
  __syncthreads();
  float s = 0.f;
  for (int i = threadIdx.x; i < L_; i += blockDim.x) { float e = __expf(v[i] - m); v[i] = e; s += e; }
  s = wave_sum(s);
  if ((threadIdx.x & 31) == 0) red[wid] = s;
  __syncthreads();
  s = 0.f;
  for (int w = 0; w < 8; ++w) s += red[w];
  const float inv = 1.0f / s;
  for (int i = threadIdx.x; i < L_; i += blockDim.x) col[(size_t)i * (B_ * H_)] = v[i] * inv;
}

// K6 (streams ebp): Amat[b,h,d] = sum_l w[l,b,h] * ebp[l,b,d]
// 2 waves per block cover a 32-float d-slab -> full 128B cacheline use in WGP$.
__global__ void __launch_bounds__(64) k_amat(const float* __restrict__ sc,
                                             const float* __restrict__ ebp,
                                             float* __restrict__ Amat) {
  const int d0 = blockIdx.x * 32 + (threadIdx.x >> 5) * 16;
  const int b  = blockIdx.y;
  wmma_tile<true, false>(sc + (size_t)b * H_, B_ * H_,
                         ebp + (size_t)b * D_ + d0, B_ * D_,
                         Amat + (size_t)b * (H_ * D_) + d0, D_,
                         L_, 1.0f, H_, 16);
}

// K7: head[b, h*32+dh] = sum_d Amat[b,h,d] * Wv1[d, h*32+dh]
__global__ void k_head(const float* __restrict__ Amat, const float* __restrict__ Wv1,
                       float* __restrict__ head) {
  const int b0 = blockIdx.x * 16, n0 = blockIdx.y * 16, h = n0 / DH_;
  wmma_tile<false, false>(Amat + (size_t)b0 * (H_ * D_) + h * D_, H_ * D_,
                          Wv1 + n0, D_,
                          head + (size_t)b0 * D_ + n0, D_,
                          D_, 1.0f, 16, 16);
}

// K11 (streams ebp): fs(l,b)=ebp[l,b,:]·p2[b,:]; lp=tanh(fs)*10 masked; softmax over L.
__global__ void k_final(const float* __restrict__ ebp, const float* __restrict__ p2,
                        const int* __restrict__ pa, float* __restrict__ out) {
  const int b = blockIdx.x;
  __shared__ float lp[L_];
  __shared__ float q[D_];
  __shared__ float red[8];
  for (int i = threadIdx.x; i < D_; i += blockDim.x) q[i] = p2[(size_t)b * D_ + i];
  __syncthreads();
  const int wid = threadIdx.x >> 5, lane = threadIdx.x & 31, nw = blockDim.x >> 5;
  for (int l = wid; l < L_; l += nw) {
    const float* row = ebp + (size_t)l * (B_ * D_) + (size_t)b * D_;
    float s = 0.f;
    for (int j = lane; j < D_; j += 32) s += row[j] * q[j];
    s = wave_sum(s);
    if (lane == 0) lp[l] = tanhf(s) * 10.0f;
  }
  __syncthreads();
  for (int t = threadIdx.x; t < T_; t += blockDim.x) lp[pa[b * T_ + t]] = -__builtin_inff();
  __syncthreads();
  float m = -__builtin_inff();
  for (int i = threadIdx.x; i < L_; i += blockDim.x) m = fmaxf(m, lp[i]);
  m = wave_max(m);
  if (lane == 0) red[wid] = m;
  __syncthreads();
  m = red[0];
  for (int w = 1; w < 8; ++w) m = fmaxf(m, red[w]);
  __syncthreads();
  float s = 0.f;
  for (int i = threadIdx.x; i < L_; i += blockDim.x) { float e = __expf(lp[i] - m); lp[i] = e; s += e; }
  s = wave_sum(s);
  if (lane == 0) red[wid] = s;
  __syncthreads();
  s = 0.f;
  for (int w = 0; w < 8; ++w) s += red[w];
  const float inv = 1.0f / s;
  for (int i = threadIdx.x; i < L_; i += blockDim.x) out[(size_t)i * B_ + b] = lp[i] * inv;
}

// ---------------------------------------------------------------------------
extern "C" void kernel_launch(void* const* d_in, const int* in_sizes, int n_in,
                              void* d_out, int out_size, void* d_ws, size_t ws_size,
                              hipStream_t stream) {
  const float* ebp = (const float*)d_in[0];
  const int*   pa  = (const int*)d_in[1];
  const float* Wq1 = (const float*)d_in[2];
  const float* Wk1 = (const float*)d_in[3];
  const float* Wv1 = (const float*)d_in[4];
  const float* Wo1 = (const float*)d_in[5];   // (H,DH,D) == flat (256,256)
  const float* Wq2 = (const float*)d_in[6];
  const float* Wk2 = (const float*)d_in[7];
  float* out = (float*)d_out;

  float* w       = (float*)d_ws;
  float* psum    = w; w += 8 * B_ * D_;           // 524288
  float* context = w; w += B_ * 3 * D_;           // 196608
  float* q1      = w; w += B_ * D_;
  float* P1      = w; w += B_ * H_ * D_;          // 524288
  float* sc      = w; w += (size_t)L_ * B_ * H_;  // 2M floats
  float* Amat    = w; w += B_ * H_ * D_;
  float* head    = w; w += B_ * D_;
  float* ctx2    = w; w += B_ * D_;
  float* q2      = w; w += B_ * D_;
  float* p2      = w; w += B_ * D_;

  // Pass 1 over ebp: split mean + gathers -> context
  k_mean_partial<<<dim3(8, B_), D_, 0, stream>>>(ebp, psum);
  k_context_fin<<<B_, D_, 0, stream>>>(ebp, psum, pa, context);
  // q1 = context(256x768) @ Wq1(768x256)
  k_gemm_nn<<<dim3(B_ / 16, D_ / 16), 32, 0, stream>>>(context, 3 * D_, Wq1, D_, q1, D_, 3 * D_, 1.0f);
  // P1 = per-head Wk1-projected query, scaled by 1/sqrt(DH)
  k_p1<<<dim3(B_ / 16, D_ / 16, H_), 32, 0, stream>>>(q1, Wk1, P1);
  // Pass 2 over ebp: attention scores (LDS-staged, async copies)
  k_scores<<<dim3(L_ / SC_LROWS, B_), 256, 0, stream>>>(ebp, P1, sc);
  // masked softmax over L per (b,h)
  k_softmax_sc<<<dim3(B_, H_), 256, 0, stream>>>(sc, pa);
  // Pass 3 over ebp: attention-weighted sum of ebp rows
  k_amat<<<dim3(D_ / 32, B_), 64, 0, stream>>>(sc, ebp, Amat);
  // head_out, context2, q2, p2  (small 256^3 WMMA GEMMs)
  k_head<<<dim3(B_ / 16, D_ / 16), 32, 0, stream>>>(Amat, Wv1, head);
  k_gemm_nn<<<dim3(B_ / 16, D_ / 16), 32, 0, stream>>>(head, D_, Wo1, D_, ctx2, D_, D_, 1.0f);
  k_gemm_nn<<<dim3(B_ / 16, D_ / 16), 32, 0, stream>>>(ctx2, D_, Wq2, D_, q2, D_, D_, 1.0f);
  k_gemm_nt<<<dim3(B_ / 16, D_ / 16), 32, 0, stream>>>(q2, D_, Wk2, D_, p2, D_, D_, 1.0f / 16.0f);
  // Pass 4 over ebp: final logits + masked softmax over L
  k_final<<<B_, 256, 0, stream>>>(ebp, p2, pa, out);
  (void)in_sizes; (void)n_in; (void)out_size; (void)ws_size;
}